// EdgeFeatures_55576876810356
// MI455X (gfx1250) — compile-verified
//
#include <hip/hip_runtime.h>

typedef float v2f __attribute__((ext_vector_type(2)));
typedef float v8f __attribute__((ext_vector_type(8)));

#define N_NB   32
#define F_IN   5
#define EF     32
#define FEAT   256
#define CUTOFF 3.0f
#define PI_F   3.14159265358979323846f

// One block (256 threads = 8 wave32) per electron n.
__global__ __launch_bounds__(256) void edge_feat_wmma_kernel(
    const float* __restrict__ r,        // [N,3]
    const float* __restrict__ r_nb,     // [N,32,3]
    const float* __restrict__ fk,       // [N,5,32]   filter_kernel
    const float* __restrict__ fb,       // [N,32]     filter_bias
    const float* __restrict__ Wg,       // [32,256]   W_gamma
    const float* __restrict__ ek,       // [N,5,256]  edge_kernel
    const float* __restrict__ eb,       // [N,256]    edge_bias
    const int*   __restrict__ s,        // [N]
    const int*   __restrict__ s_nb,     // [N,32]
    float* __restrict__ out_gamma,      // [N,32,256]
    float* __restrict__ out_edge)       // [N,32,256]
{
    // Row stride 264 dwords: the two K-half reads (lanes 0-15 vs 16-31, rows k
    // and k+2) differ by 528 dwords == 16 mod 64 banks -> disjoint bank ranges.
    __shared__ float s_w[EF][264];      // staged W_gamma, ~33 KB
    __shared__ float s_beta[N_NB][36];  // padded for A-fragment b64 loads
    __shared__ float s_feats[N_NB][8];
    __shared__ float s_env[N_NB];

    const int n = blockIdx.x;
    const int t = threadIdx.x;

    // ---- Phase 0: stage W_gamma into LDS (coalesced) ----
    for (int i = t; i < EF * FEAT; i += 256) {
        s_w[i >> 8][i & 255] = Wg[i];
    }

    // ---- Phase 1: feats + cutoff envelope (32 lanes) ----
    if (t < N_NB) {
        const float rx = r[n * 3 + 0], ry = r[n * 3 + 1], rz = r[n * 3 + 2];
        const float dx = r_nb[(n * N_NB + t) * 3 + 0] - rx;
        const float dy = r_nb[(n * N_NB + t) * 3 + 1] - ry;
        const float dz = r_nb[(n * N_NB + t) * 3 + 2] - rz;
        const float dist = sqrtf(dx * dx + dy * dy + dz * dz);
        const float same = (s[n] == s_nb[n * N_NB + t]) ? 1.0f : 0.0f;
        s_feats[t][0] = dx;
        s_feats[t][1] = dy;
        s_feats[t][2] = dz;
        s_feats[t][3] = dist;
        s_feats[t][4] = same;
        float x = dist * (1.0f / CUTOFF);
        x = fminf(fmaxf(x, 0.0f), 1.0f);
        s_env[t] = (dist < CUTOFF) ? 0.5f * (cosf(PI_F * x) + 1.0f) : 0.0f;
    }
    __syncthreads();

    // ---- Phase 2: beta[32][32] = env * (feats @ filter_kernel + filter_bias) ----
    {
        const int k  = t >> 3;            // neighbor row 0..31
        const int e0 = (t & 7) * 4;       // 4 edge-feature columns per thread
        const float env = s_env[k];
        const float f0 = s_feats[k][0], f1 = s_feats[k][1], f2 = s_feats[k][2];
        const float f3 = s_feats[k][3], f4 = s_feats[k][4];
        const float* fkn = fk + (size_t)n * F_IN * EF;
        const float* fbn = fb + (size_t)n * EF;
#pragma unroll
        for (int j = 0; j < 4; ++j) {
            const int e = e0 + j;
            float acc = fbn[e];
            acc = fmaf(f0, fkn[0 * EF + e], acc);
            acc = fmaf(f1, fkn[1 * EF + e], acc);
            acc = fmaf(f2, fkn[2 * EF + e], acc);
            acc = fmaf(f3, fkn[3 * EF + e], acc);
            acc = fmaf(f4, fkn[4 * EF + e], acc);
            s_beta[k][e] = env * acc;
        }
    }

    // ---- Phase 3: edge_features = feats @ edge_kernel + edge_bias ----
    // Column-per-thread: stores are fully coalesced (wave covers 32 consecutive cols).
    {
        const int e = t;
        const float* ekn = ek + (size_t)n * F_IN * FEAT;
        const float k0 = ekn[0 * FEAT + e], k1 = ekn[1 * FEAT + e];
        const float k2 = ekn[2 * FEAT + e], k3 = ekn[3 * FEAT + e];
        const float k4 = ekn[4 * FEAT + e];
        const float bias = eb[(size_t)n * FEAT + e];
        float* dst = out_edge + (size_t)n * N_NB * FEAT + e;
#pragma unroll
        for (int k = 0; k < N_NB; ++k) {
            float v = bias;
            v = fmaf(k0, s_feats[k][0], v);
            v = fmaf(k1, s_feats[k][1], v);
            v = fmaf(k2, s_feats[k][2], v);
            v = fmaf(k3, s_feats[k][3], v);
            v = fmaf(k4, s_feats[k][4], v);
            dst[k * FEAT] = v;
        }
    }
    __syncthreads();

    // ---- Phase 4: Gamma[32][256] = beta[32][32] @ W_gamma[32][256] via fp32 WMMA ----
    {
        const int w     = t >> 5;        // wave 0..7 -> N-tile pair {2w, 2w+1}
        const int lane  = t & 31;
        const int lhalf = lane >> 4;     // 0: K / rows 0-7 half, 1: K+2 / rows 8-15 half
        const int lmod  = lane & 15;
        float* g = out_gamma + (size_t)n * N_NB * FEAT;
#pragma unroll
        for (int mt = 0; mt < 2; ++mt) {
#pragma unroll
            for (int ntl = 0; ntl < 2; ++ntl) {
                const int nt  = 2 * w + ntl;
                const int col = nt * 16 + lmod;
                v8f acc = {};
#pragma unroll
                for (int ks = 0; ks < 8; ++ks) {
                    const int kb = 4 * ks + 2 * lhalf;
                    // A fragment: 16x4 f32 (lanes 0-15: K,K+1; lanes 16-31: K+2,K+3)
                    v2f a = *(const v2f*)&s_beta[mt * 16 + lmod][kb];
                    // B fragment: 4x16 f32, row-striped across lanes
                    v2f b;
                    b.x = s_w[kb + 0][col];
                    b.y = s_w[kb + 1][col];
                    acc = __builtin_amdgcn_wmma_f32_16x16x4_f32(
                        false, a, false, b, (short)0, acc, false, false);
                }
                // D layout: VGPR r -> row r (lanes 0-15) / row 8+r (lanes 16-31)
                const int rbase = mt * 16 + lhalf * 8;
#pragma unroll
                for (int rr = 0; rr < 8; ++rr) {
                    g[(rbase + rr) * FEAT + col] = acc[rr];
                }
            }
        }
    }
}

extern "C" void kernel_launch(void* const* d_in, const int* in_sizes, int n_in,
                              void* d_out, int out_size, void* d_ws, size_t ws_size,
                              hipStream_t stream) {
    const float* r    = (const float*)d_in[0];
    const float* r_nb = (const float*)d_in[1];
    const float* fk   = (const float*)d_in[2];
    const float* fb   = (const float*)d_in[3];
    const float* Wg   = (const float*)d_in[4];
    const float* ek   = (const float*)d_in[5];
    const float* eb   = (const float*)d_in[6];
    const int*   s    = (const int*)d_in[7];
    const int*   s_nb = (const int*)d_in[8];

    const int n_el = in_sizes[0] / 3;   // r is [N,3]
    float* out_gamma = (float*)d_out;
    float* out_edge  = out_gamma + (size_t)n_el * N_NB * FEAT;

    edge_feat_wmma_kernel<<<n_el, 256, 0, stream>>>(
        r, r_nb, fk, fb, Wg, ek, eb, s, s_nb, out_gamma, out_edge);
}